// PointGroup_28303834481491
// MI455X (gfx1250) — compile-verified
//
#include <hip/hip_runtime.h>
#include <hip/hip_bf16.h>

#define N_PTS   262144
#define C_DIM   32
#define NCLUSTR 4096
#define SUMNP   524288
#define BB      4
#define PP      2048
#define KNN     8
#define THR2    0.09f   /* 0.3^2 : compare squared distances */

typedef __attribute__((ext_vector_type(2))) float v2f;
typedef __attribute__((ext_vector_type(8))) float v8f;

// ---- monotonic float<->uint encoding for atomic max ----
__device__ __forceinline__ unsigned enc_f32(float x) {
    unsigned b = __float_as_uint(x);
    return (b & 0x80000000u) ? ~b : (b | 0x80000000u);
}
__device__ __forceinline__ float dec_f32(unsigned e) {
    unsigned b = (e & 0x80000000u) ? (e & 0x7fffffffu) : ~e;
    return __uint_as_float(b);
}

// ---------------- cluster feature max-pool ----------------
__global__ void pg_init_enc(unsigned* enc_buf, int n) {
    int i = blockIdx.x * blockDim.x + threadIdx.x;
    if (i < n) enc_buf[i] = 0u;   // sentinel -> finalized to 0
}

// cluster_ids are sorted: fold 8 consecutive pairs per thread, atomic only on
// segment boundaries (avg segment length = SUMNP/NCLUST = 128 -> ~8x fewer atomics)
__global__ void pg_scatter_max(const float* __restrict__ feats,
                               const int* __restrict__ cluster_ids,
                               const int* __restrict__ point_ids,
                               unsigned* enc_buf) {
    int tid = blockIdx.x * blockDim.x + threadIdx.x;   // (SUMNP/8)*32 threads
    int c   = tid & 31;
    int s0  = (tid >> 5) * 8;
    if (s0 >= SUMNP) return;
    int cur_id = cluster_ids[s0];
    unsigned acc = 0u;
#pragma unroll
    for (int t = 0; t < 8; t++) {
        int s  = s0 + t;
        int cid = cluster_ids[s];
        unsigned ev = enc_f32(feats[point_ids[s] * C_DIM + c]);
        if (cid != cur_id) {
            atomicMax(&enc_buf[cur_id * C_DIM + c], acc);
            cur_id = cid;
            acc = ev;
        } else {
            acc = (acc > ev) ? acc : ev;
        }
    }
    atomicMax(&enc_buf[cur_id * C_DIM + c], acc);
}

__global__ void pg_finalize(unsigned* enc_buf, float* out, int n) {
    int i = blockIdx.x * blockDim.x + threadIdx.x;
    if (i >= n) return;
    unsigned e = enc_buf[i];
    out[i] = (e == 0u) ? 0.0f : dec_f32(e);
}

// ---------------- kNN-8 on shifted centers (squared distances) -------------
__global__ void pg_knn(const float* __restrict__ centers,
                       const float* __restrict__ offs,
                       int* __restrict__ knn_out) {
    const int row_g = blockIdx.x;        // 0 .. BB*PP-1
    const int b     = row_g / PP;
    const int row   = row_g - b * PP;
    const int base  = b * PP;
    const int tid   = threadIdx.x;       // 128 threads

    __shared__ float s_d[128 * KNN];
    __shared__ int   s_i[128 * KNN];

    float rx = centers[row_g * 3 + 0] + offs[row_g * 3 + 0];
    float ry = centers[row_g * 3 + 1] + offs[row_g * 3 + 1];
    float rz = centers[row_g * 3 + 2] + offs[row_g * 3 + 2];

    float bd[KNN]; int bi[KNN];
#pragma unroll
    for (int k = 0; k < KNN; k++) { bd[k] = 3.0e38f; bi[k] = row; }

    for (int j = tid; j < PP; j += 128) {
        int g = (base + j) * 3;
        float dx = centers[g + 0] + offs[g + 0] - rx;
        float dy = centers[g + 1] + offs[g + 1] - ry;
        float dz = centers[g + 2] + offs[g + 2] - rz;
        float d2 = dx * dx + dy * dy + dz * dz;
        if (d2 < bd[KNN - 1]) {
            int k = KNN - 1;
            while (k > 0 && bd[k - 1] > d2) { bd[k] = bd[k - 1]; bi[k] = bi[k - 1]; k--; }
            bd[k] = d2; bi[k] = j;
        }
    }
#pragma unroll
    for (int k = 0; k < KNN; k++) { s_d[tid * KNN + k] = bd[k]; s_i[tid * KNN + k] = bi[k]; }
    __syncthreads();

    for (int stride = 64; stride >= 1; stride >>= 1) {
        if (tid < stride) {
            float md[KNN]; int mi[KNN];
            float* A  = &s_d[tid * KNN];            int* Ai = &s_i[tid * KNN];
            float* Bd = &s_d[(tid + stride) * KNN]; int* Bi = &s_i[(tid + stride) * KNN];
            int a = 0, c = 0;
#pragma unroll
            for (int k = 0; k < KNN; k++) {
                if (A[a] <= Bd[c]) { md[k] = A[a]; mi[k] = Ai[a]; a++; }
                else               { md[k] = Bd[c]; mi[k] = Bi[c]; c++; }
            }
#pragma unroll
            for (int k = 0; k < KNN; k++) { A[k] = md[k]; Ai[k] = mi[k]; }
        }
        __syncthreads();
    }

    if (tid < KNN) {
        float dd2 = s_d[tid];
        int   id  = s_i[tid];
        knn_out[row_g * KNN + tid] = (dd2 < THR2) ? id : row;   // safe_idx semantics
    }
}

// ---------------- WMMA 16x16 adjacency tiles (LDS-free, branchless) --------
__global__ void __launch_bounds__(32)
pg_graph_tile(const float* __restrict__ centers,
              const float* __restrict__ offs,
              const int* __restrict__ sem,
              const int* __restrict__ knn,
              float* __restrict__ adj_out) {
    const int b  = blockIdx.z;
    const int i0 = blockIdx.y * 16;
    const int j0 = blockIdx.x * 16;
    const int lane = threadIdx.x;       // wave32, EXEC all-ones for WMMA
    const int hl = lane & 15;
    const int hi = lane >> 4;

    const int gi = b * PP + i0 + hl;    // row-side overseg held by this lane
    const int gj = b * PP + j0 + hl;    // col-side overseg held by this lane

    float cxi = centers[gi * 3 + 0], cyi = centers[gi * 3 + 1], czi = centers[gi * 3 + 2];
    float sxi = cxi + offs[gi * 3 + 0], syi = cyi + offs[gi * 3 + 1], szi = czi + offs[gi * 3 + 2];
    float cxj = centers[gj * 3 + 0], cyj = centers[gj * 3 + 1], czj = centers[gj * 3 + 2];
    float sxj = cxj + offs[gj * 3 + 0], syj = cyj + offs[gj * 3 + 1], szj = czj + offs[gj * 3 + 2];

    float rcn = cxi * cxi + cyi * cyi + czi * czi;   // |center_i|^2 (row hl)
    float rsn = sxi * sxi + syi * syi + szi * szi;   // |shift_i|^2
    float ccn = cxj * cxj + cyj * cyj + czj * czj;   // |center_j|^2 (col hl)
    float csn = sxj * sxj + syj * syj + szj * szj;
    int   rsem = sem[gi];
    int   csem = sem[gj];

    // 16-bit in-tile kNN membership masks (branch-free build)
    const int4* kr = (const int4*)(knn + gi * KNN);   // 32B-aligned
    const int4* kc = (const int4*)(knn + gj * KNN);
    int4 kr0 = kr[0], kr1 = kr[1];
    int4 kc0 = kc[0], kc1 = kc[1];
    int rk[KNN] = { kr0.x, kr0.y, kr0.z, kr0.w, kr1.x, kr1.y, kr1.z, kr1.w };
    int ck[KNN] = { kc0.x, kc0.y, kc0.z, kc0.w, kc1.x, kc1.y, kc1.z, kc1.w };
    int rowmask = 0, colmask = 0;
#pragma unroll
    for (int k = 0; k < KNN; k++) {
        int t = rk[k] - j0;   // column hit within tile?
        rowmask |= ((unsigned)t < 16u) ? (1 << (t & 15)) : 0;
        int u = ck[k] - i0;   // row hit within tile?
        colmask |= ((unsigned)u < 16u) ? (1 << (u & 15)) : 0;
    }
    // pack row-side semantic (<32, fits 8 bits) with rowmask: one shuffle, not two
    int rpack = (rowmask << 8) | (rsem & 0xFF);

    // A fragment (16x4 f32): lanes 0-15 hold K=0,1 ; lanes 16-31 hold K=2,3 (K=3 pad 0)
    v2f ac, bc, as_, bs;
    ac.x = hi ? czi : cxi;  ac.y = hi ? 0.0f : cyi;
    bc.x = hi ? czj : cxj;  bc.y = hi ? 0.0f : cyj;
    as_.x = hi ? szi : sxi; as_.y = hi ? 0.0f : syi;
    bs.x = hi ? szj : sxj;  bs.y = hi ? 0.0f : syj;

    v8f zero = {};
    v8f dotc = __builtin_amdgcn_wmma_f32_16x16x4_f32(false, ac, false, bc,
                                                     (short)0, zero, false, false);
    v8f dots = __builtin_amdgcn_wmma_f32_16x16x4_f32(false, as_, false, bs,
                                                     (short)0, zero, false, false);

    float* batch_out = adj_out + (size_t)b * PP * PP;
#pragma unroll
    for (int r = 0; r < 8; r++) {
        int m = r + 8 * hi;                 // C/D layout: lanes 0-15 -> M=r, 16-31 -> M=r+8
        // row-side values live in lane m's registers (3 ds_bpermute per r)
        float rcn_m   = __shfl(rcn, m);
        float rsn_m   = __shfl(rsn, m);
        int   rpack_m = __shfl(rpack, m);
        int   rsem_m  = rpack_m & 0xFF;
        int   rmsk_m  = rpack_m >> 8;
        float d2c = fmaxf(rcn_m + ccn - 2.0f * dotc[r], 0.0f);
        float d2s = fmaxf(rsn_m + csn - 2.0f * dots[r], 0.0f);
        int squeeze = (d2s < d2c) ? 1 : 0;   // sqrt is monotonic
        int li = i0 + m;
        int lj = j0 + hl;
        int conn = ((li == lj) ? 1 : 0) |
                   ((((rsem_m == csem) ? 1 : 0) & ((rsem_m >= 2) ? 1 : 0)));
        int edge = ((rmsk_m >> hl) & 1) | ((colmask >> m) & 1);
        batch_out[(size_t)li * PP + lj] = (float)(edge & conn & squeeze);
    }
}

// ---------------- connect map (float4 stores) ----------------
__global__ void pg_connect(const int* __restrict__ sem, float4* __restrict__ out) {
    int idx = blockIdx.x * blockDim.x + threadIdx.x;   // BB*PP*PP/4 threads
    const int rowq = PP / 4;
    int b   = idx / (PP * rowq);
    int rem = idx - b * PP * rowq;
    int i   = rem / rowq;
    int j0  = (rem - i * rowq) * 4;
    int si  = sem[b * PP + i];
    float4 v;
    {
        int sj0 = sem[b * PP + j0 + 0];
        int sj1 = sem[b * PP + j0 + 1];
        int sj2 = sem[b * PP + j0 + 2];
        int sj3 = sem[b * PP + j0 + 3];
        int c0 = ((i == j0 + 0) ? 1 : 0) | (((si == sj0) ? 1 : 0) & ((si >= 2) ? 1 : 0));
        int c1 = ((i == j0 + 1) ? 1 : 0) | (((si == sj1) ? 1 : 0) & ((si >= 2) ? 1 : 0));
        int c2 = ((i == j0 + 2) ? 1 : 0) | (((si == sj2) ? 1 : 0) & ((si >= 2) ? 1 : 0));
        int c3 = ((i == j0 + 3) ? 1 : 0) | (((si == sj3) ? 1 : 0) & ((si >= 2) ? 1 : 0));
        v.x = (float)c0; v.y = (float)c1; v.z = (float)c2; v.w = (float)c3;
    }
    out[idx] = v;
}

extern "C" void kernel_launch(void* const* d_in, const int* in_sizes, int n_in,
                              void* d_out, int out_size, void* d_ws, size_t ws_size,
                              hipStream_t stream) {
    (void)in_sizes; (void)n_in; (void)out_size; (void)d_ws; (void)ws_size;

    const float* feats       = (const float*)d_in[0];
    const int*   cluster_ids = (const int*)  d_in[1];
    const int*   point_ids   = (const int*)  d_in[2];
    const float* centers     = (const float*)d_in[3];
    const float* offs        = (const float*)d_in[4];
    const int*   sem         = (const int*)  d_in[5];

    float* out         = (float*)d_out;
    float* out_cluster = out;                                   // 131072
    float* out_adj     = out + 131072;                          // 16777216
    float* out_conn    = out + 131072 + (size_t)BB * PP * PP;   // 16777216

    unsigned* enc_buf = (unsigned*)out_cluster;
    int* knn_buf = (int*)out_conn;   // staged here, overwritten by pg_connect last

    const int nclf = NCLUSTR * C_DIM;   // 131072

    // 1) cluster max-pool (sorted-run folding + atomic u32 max)
    pg_init_enc<<<(nclf + 255) / 256, 256, 0, stream>>>(enc_buf, nclf);
    pg_scatter_max<<<((SUMNP / 8) * 32) / 256, 256, 0, stream>>>(feats, cluster_ids,
                                                                 point_ids, enc_buf);
    pg_finalize<<<(nclf + 255) / 256, 256, 0, stream>>>(enc_buf, out_cluster, nclf);

    // 2) kNN on shifts (one block per overseg row)
    pg_knn<<<BB * PP, 128, 0, stream>>>(centers, offs, knn_buf);

    // 3) adjacency tiles via v_wmma_f32_16x16x4_f32 (LDS-free, shuffle-based)
    dim3 g(PP / 16, PP / 16, BB);
    pg_graph_tile<<<g, 32, 0, stream>>>(centers, offs, sem, knn_buf, out_adj);

    // 4) connect map (overwrites knn staging area)
    pg_connect<<<(BB * PP * (PP / 4)) / 256, 256, 0, stream>>>(sem, (float4*)out_conn);
}